// FixPoint_21861383536880
// MI455X (gfx1250) — compile-verified
//
#include <hip/hip_runtime.h>
#include <cstdint>

#define FDIM 64
#define MAX_ITERS 48       // convergence expected ~iter 20 (alpha=0.5 contraction); 48 = safety margin
#define NPB 8              // nodes per block (1 wave32 per node)
#define TPB 256            // 8 waves
#define LDS_EDGE_CAP 2048  // float2 entries = 16 KB; expected block edge window ~128 (Poisson)
#define ALPHA 0.5f
#define ATOL 1e-5f
#define RTOL 1e-5f

// One damped Jacobi step: xout = 0.5*A_hat*xin + 0.5*b, with the reference stopping criterion.
// Edge metadata for the block's contiguous CSR window is staged via CDNA5 async-to-LDS DMA
// (ASYNCcnt-tracked), then broadcast-read from LDS by all lanes of each node's wave.
__global__ __launch_bounds__(TPB) void step_kernel(int k, const float* __restrict__ xin,
    float* __restrict__ xout, const float* __restrict__ bsrc,
    const int* __restrict__ row_ptr, const float2* __restrict__ wc,
    int* __restrict__ flags, int n) {
  __shared__ float2 lds_wc[LDS_EDGE_CAP];
  __shared__ int viol;
  const int tid  = threadIdx.x;
  const int wave = tid >> 5;
  const int lane = tid & 31;
  const int n0 = blockIdx.x * NPB;
  const int n1 = (n0 + NPB < n) ? (n0 + NPB) : n;
  const int node = n0 + wave;

  // Convergence gating with exact reference semantics:
  //   flags[j]==0  => step j found x_old ~= x_new (allclose) => x_{j+1} is the answer.
  //   first step after convergence copies (syncs both ping-pong buffers);
  //   every later step is a pure no-op (both buffers already hold the fixed point).
  const bool prev0 = (k >= 1) && (flags[k - 1] == 0);
  const bool prev1 = (k >= 2) && (flags[k - 2] == 0);
  if (prev0 && prev1) return;            // dead step: zero traffic
  if (prev0) {                           // first post-convergence step: buffer sync copy
    if (node < n1) {
      int off = node * FDIM + lane * 2;
      float2 v = *(const float2*)(xin + off);
      *(float2*)(xout + off) = v;
    }
    return;
  }

  if (tid == 0) viol = 0;
  int ebase = row_ptr[n0];
  int etot  = row_ptr[n1] - ebase;
  bool use_lds = (etot <= LDS_EDGE_CAP);
  if (use_lds) {
    // CDNA5 async DMA: per-lane global -> LDS, no VGPR round trip, tracked by ASYNCcnt.
    for (int i = tid; i < etot; i += TPB) {
      unsigned loff = (unsigned)(uintptr_t)(void*)(&lds_wc[i]);
      unsigned long long ga = (unsigned long long)(uintptr_t)(const void*)(wc + ebase + i);
      asm volatile("global_load_async_to_lds_b64 %0, %1, off"
                   :: "v"(loff), "v"(ga) : "memory");
    }
    asm volatile("s_wait_asynccnt 0" ::: "memory");
  }
  __syncthreads();

  if (node < n1) {
    int s  = row_ptr[node] - ebase;
    int e2 = row_ptr[node + 1] - ebase;
    float2 acc = make_float2(0.f, 0.f);
    if (use_lds) {
      for (int j = s; j < e2; ++j) {
        float2 p = lds_wc[j];                 // LDS broadcast: all 32 lanes same address
        int cix = __float_as_int(p.y);
        float2 xv = *(const float2*)(xin + cix * FDIM + lane * 2);  // coalesced 256B row gather (L2-resident)
        acc.x = fmaf(p.x, xv.x, acc.x);
        acc.y = fmaf(p.x, xv.y, acc.y);
      }
    } else {
      for (int j = s; j < e2; ++j) {
        float2 p = wc[ebase + j];
        int cix = __float_as_int(p.y);
        float2 xv = *(const float2*)(xin + cix * FDIM + lane * 2);
        acc.x = fmaf(p.x, xv.x, acc.x);
        acc.y = fmaf(p.x, xv.y, acc.y);
      }
    }
    int off = node * FDIM + lane * 2;
    float2 bo = *(const float2*)(bsrc + off);
    float2 xo = *(const float2*)(xin + off);
    float2 xn;
    xn.x = ALPHA * acc.x + (1.0f - ALPHA) * bo.x;
    xn.y = ALPHA * acc.y + (1.0f - ALPHA) * bo.y;
    *(float2*)(xout + off) = xn;
    bool bad = (fabsf(xo.x - xn.x) > ATOL + RTOL * fabsf(xn.x)) ||
               (fabsf(xo.y - xn.y) > ATOL + RTOL * fabsf(xn.y));
    if (bad) viol = 1;   // many writers, same value: benign
  }
  __syncthreads();
  if (tid == 0 && viol) flags[k] = 1;
}

__global__ void init_kernel(float* deg, int* cnt, int* flags, int n) {
  int i = blockIdx.x * blockDim.x + threadIdx.x;
  if (i < n) { deg[i] = 0.0f; cnt[i] = 0; }
  if (i < MAX_ITERS) flags[i] = 0;
}

__global__ void degcnt_kernel(const float* __restrict__ ew, const int* __restrict__ dst,
                              float* deg, int* cnt, int E) {
  int e = blockIdx.x * blockDim.x + threadIdx.x;
  if (e < E) {
    int d = dst[e];
    atomicAdd(&deg[d], ew[e]);
    atomicAdd(&cnt[d], 1);
  }
}

// ---- 3-phase parallel exclusive scan of cnt -> row_ptr (and cursor prime) ----
__global__ __launch_bounds__(1024) void scanA_kernel(const int* __restrict__ cnt, int* bsum, int n) {
  __shared__ int sdata[1024];
  int tid = threadIdx.x;
  int i = blockIdx.x * 1024 + tid;
  sdata[tid] = (i < n) ? cnt[i] : 0;
  __syncthreads();
  for (int off = 512; off > 0; off >>= 1) {
    if (tid < off) sdata[tid] += sdata[tid + off];
    __syncthreads();
  }
  if (tid == 0) bsum[blockIdx.x] = sdata[0];
}

__global__ __launch_bounds__(1024) void scanB_kernel(int* bsum, int* row_ptr, int nb, int n) {
  __shared__ int sdata[1024];
  int tid = threadIdx.x;
  int v = (tid < nb) ? bsum[tid] : 0;
  sdata[tid] = v;
  __syncthreads();
  for (int off = 1; off < 1024; off <<= 1) {
    int t = (tid >= off) ? sdata[tid - off] : 0;
    __syncthreads();
    sdata[tid] += t;
    __syncthreads();
  }
  if (tid < nb) bsum[tid] = sdata[tid] - v;     // exclusive block offsets
  if (tid == 1023) row_ptr[n] = sdata[1023];    // grand total = E
}

__global__ __launch_bounds__(1024) void scanC_kernel(const int* __restrict__ cnt,
                                                     const int* __restrict__ bsum,
                                                     int* row_ptr, int* cursor, int n) {
  __shared__ int sdata[1024];
  int tid = threadIdx.x;
  int i = blockIdx.x * 1024 + tid;
  int v = (i < n) ? cnt[i] : 0;
  sdata[tid] = v;
  __syncthreads();
  for (int off = 1; off < 1024; off <<= 1) {
    int t = (tid >= off) ? sdata[tid - off] : 0;
    __syncthreads();
    sdata[tid] += t;
    __syncthreads();
  }
  if (i < n) {
    int rp = bsum[blockIdx.x] + sdata[tid] - v;
    row_ptr[i] = rp;
    cursor[i] = rp;
  }
}

// CSR fill with folded dst-stochastic normalization; packed (w, src) records.
__global__ void fill_kernel(const float* __restrict__ ew, const int* __restrict__ src,
                            const int* __restrict__ dst, const float* __restrict__ deg,
                            int* cursor, float2* wc, int E) {
  int e = blockIdx.x * blockDim.x + threadIdx.x;
  if (e < E) {
    int d = dst[e];
    float w = ew[e] / fmaxf(deg[d], 1e-12f);
    int pos = atomicAdd(&cursor[d], 1);
    wc[pos] = make_float2(w, __int_as_float(src[e]));
  }
}

extern "C" void kernel_launch(void* const* d_in, const int* in_sizes, int n_in,
                              void* d_out, int out_size, void* d_ws, size_t ws_size,
                              hipStream_t stream) {
  const float* x   = (const float*)d_in[0];
  const float* ew  = (const float*)d_in[1];
  const float* b   = (const float*)d_in[2];
  const int*   src = (const int*)d_in[3];
  const int*   dst = (const int*)d_in[4];
  const int N = in_sizes[0] / FDIM;
  const int E = in_sizes[1];

  char* p = (char*)d_ws;
  auto alloc = [&](size_t bytes) -> char* {
    char* r = p; p += (bytes + 255) & ~(size_t)255; return r;
  };
  const int nbscan = (N + 1023) / 1024;          // <= 1024 assumed (N up to ~1M)
  float*  deg     = (float*)alloc((size_t)N * sizeof(float));
  int*    cnt     = (int*)alloc((size_t)N * sizeof(int));
  int*    row_ptr = (int*)alloc((size_t)(N + 1) * sizeof(int));
  int*    cursor  = (int*)alloc((size_t)N * sizeof(int));
  int*    bsum    = (int*)alloc((size_t)nbscan * sizeof(int));
  float2* wc      = (float2*)alloc((size_t)E * sizeof(float2));
  float*  buf0    = (float*)alloc((size_t)N * FDIM * sizeof(float));
  int*    flags   = (int*)alloc(MAX_ITERS * sizeof(int));
  float*  buf1    = (float*)d_out;   // even MAX_ITERS -> d_out always holds the final iterate

  int ib = (N + 255) / 256;
  int eb = (E + 255) / 256;
  init_kernel<<<ib, 256, 0, stream>>>(deg, cnt, flags, N);
  degcnt_kernel<<<eb, 256, 0, stream>>>(ew, dst, deg, cnt, E);
  scanA_kernel<<<nbscan, 1024, 0, stream>>>(cnt, bsum, N);
  scanB_kernel<<<1, 1024, 0, stream>>>(bsum, row_ptr, nbscan, N);
  scanC_kernel<<<nbscan, 1024, 0, stream>>>(cnt, bsum, row_ptr, cursor, N);
  fill_kernel<<<eb, 256, 0, stream>>>(ew, src, dst, deg, cursor, wc, E);

  int nb = (N + NPB - 1) / NPB;
  float* bufs[2] = { buf0, buf1 };
  for (int k = 0; k < MAX_ITERS; ++k) {
    const float* in = (k == 0) ? x : bufs[(k - 1) & 1];
    float* out = bufs[k & 1];
    step_kernel<<<nb, TPB, 0, stream>>>(k, in, out, b, row_ptr, wc, flags, N);
  }
}